// VariantCoeLinear1d_82291573391972
// MI455X (gfx1250) — compile-verified
//
#include <hip/hip_runtime.h>
#include <hip/hip_bf16.h>
#include <stdint.h>

#define BATCH 32
#define NCELL 4096
#define STEPS 600
#define TPB   1024   // 4 cells per thread, 32 waves per workgroup

// f(u) = 0.5*u*(3-u^2) + (10/12)*u^2*(0.75 - 2u + 1.5u^2 - 0.25u^4)
__device__ __forceinline__ float f_flux(float x) {
    float x2 = x * x;
    return 0.5f * x * (3.0f - x2)
         + 0.8333333333333334f * x2 * (0.75f - 2.0f * x + 1.5f * x2 - 0.25f * x2 * x2);
}

// |f'(u)| = |1.5 + 1.25u - 6.5u^2 + 5u^3 - 1.25u^5|  (exact derivative, Horner form)
__device__ __forceinline__ float f_wavespeed(float x) {
    float d = 1.5f + x * (1.25f + x * (-6.5f + x * (5.0f - 1.25f * x * x)));
    return fabsf(d);
}

__global__ __launch_bounds__(TPB) void llf_traj_kernel(const float* __restrict__ init,
                                                       float* __restrict__ out) {
    __shared__ __align__(16) float u[2][NCELL];   // 32 KB double buffer, row lives in LDS

    const int b  = blockIdx.x;       // batch row
    const int t  = threadIdx.x;      // 0..1023
    const int c0 = t << 2;           // first of 4 cells owned by this thread
    const float dtdx = (float)(0.0009 * 4096.0 / 10.0);  // DT/DX = 0.36864

    // 32-bit byte offset of this thread's 4 floats within a row (GVS-mode vaddr)
    const unsigned voff = (unsigned)(c0 * 4);
    // raw LDS byte addresses (low 32 bits of the generic pointer == LDS offset)
    const unsigned ldsA = (unsigned)(uintptr_t)(&u[0][c0]);
    const unsigned ldsB = (unsigned)(uintptr_t)(&u[1][c0]);

    // ---- async load init row straight into LDS (memory -> LDS, no VGPR payload) ----
    const uint64_t src = (uint64_t)(init + (size_t)b * NCELL);
    asm volatile("global_load_async_to_lds_b128 %0, %1, %2"
                 :: "v"(ldsA), "v"(voff), "s"(src) : "memory");
    asm volatile("s_wait_asynccnt 0x0" ::: "memory");
    __syncthreads();

    // ---- stream slab 0 (the init state) from LDS to out[0, b, :] ----
    {
        const uint64_t dst0 = (uint64_t)(out + (size_t)b * NCELL);
        asm volatile("global_store_async_from_lds_b128 %0, %1, %2 th:TH_STORE_NT"
                     :: "v"(voff), "v"(ldsA), "s"(dst0) : "memory");
    }

    int cur = 0;
    for (int step = 0; step < STEPS; ++step) {
        const float* __restrict__ up = u[cur];

        // 6-point stencil: cells c0-1 .. c0+4 (ends clamped; clamped values only
        // feed the boundary cells, which are overwritten by the outflow BC below)
        float w[6];
        {
            float4 mid = ((const float4*)up)[t];          // ds_load_b128
            w[0] = up[(c0 == 0) ? 0 : (c0 - 1)];
            w[1] = mid.x; w[2] = mid.y; w[3] = mid.z; w[4] = mid.w;
            w[5] = up[(c0 + 4 < NCELL) ? (c0 + 4) : (NCELL - 1)];
        }

        float fv[6], av[6];
#pragma unroll
        for (int k = 0; k < 6; ++k) { fv[k] = f_flux(w[k]); av[k] = f_wavespeed(w[k]); }

        // interface fluxes fh[k] between points w[k], w[k+1]  (global interface c0-1+k)
        float fh[5];
#pragma unroll
        for (int k = 0; k < 5; ++k) {
            float M = fmaxf(av[k], av[k + 1]);
            fh[k] = 0.5f * (fv[k] + fv[k + 1]) - 0.5f * M * (w[k + 1] - w[k]);
        }

        // interior update; cell c0+i uses interfaces fh[i+1] - fh[i]
        float nu[4];
#pragma unroll
        for (int i = 0; i < 4; ++i) nu[i] = w[i + 1] - dtdx * (fh[i + 1] - fh[i]);

        // outflow BCs: u[0] = u[1], u[N-1] = u[N-2]
        if (c0 == 0)            nu[0] = nu[1];
        if (c0 + 4 == NCELL)    nu[3] = nu[2];

        // The buffer we are about to overwrite was async-stored two steps ago;
        // allow only the most recent store (last step) to remain in flight.
        asm volatile("s_wait_asynccnt 0x1" ::: "memory");

        float* __restrict__ un = u[cur ^ 1];
        ((float4*)un)[t] = make_float4(nu[0], nu[1], nu[2], nu[3]);  // ds_store_b128

        __syncthreads();                                   // all waves finished writing un
        asm volatile("s_wait_dscnt 0x0" ::: "memory");     // our LDS write landed

        // ---- stream this step's row LDS -> out[step+1, b, :] (overlaps next step) ----
        const uint64_t dst =
            (uint64_t)(out + ((size_t)(step + 1) * BATCH + (size_t)b) * NCELL);
        const unsigned ldsn = (cur == 0) ? ldsB : ldsA;
        asm volatile("global_store_async_from_lds_b128 %0, %1, %2 th:TH_STORE_NT"
                     :: "v"(voff), "v"(ldsn), "s"(dst) : "memory");

        cur ^= 1;
    }

    // drain outstanding async stores before wave teardown
    asm volatile("s_wait_asynccnt 0x0" ::: "memory");
}

extern "C" void kernel_launch(void* const* d_in, const int* in_sizes, int n_in,
                              void* d_out, int out_size, void* d_ws, size_t ws_size,
                              hipStream_t stream) {
    (void)in_sizes; (void)n_in; (void)out_size; (void)d_ws; (void)ws_size;
    const float* init = (const float*)d_in[0];   // (32, 4096) f32
    // d_in[1] is stepnum (==600, fixed per reference)
    float* out = (float*)d_out;                  // (601, 32, 4096) f32

    llf_traj_kernel<<<dim3(BATCH), dim3(TPB), 0, stream>>>(init, out);
}